// GGNN_model_14242111553875
// MI455X (gfx1250) — compile-verified
//
#include <hip/hip_runtime.h>
#include <hip/hip_bf16.h>
#include <math.h>

// ---------------------------------------------------------------------------
// GGNN: B=512 graphs, N=1024 nodes, C=15 (pad 16), E=16384 edges, L=3 layers,
// GRU cell, mean pool, MLP 15->256->256->1.
// Strategy: one workgroup per graph; h/agg resident in LDS; LDS ds_add_f32
// scatter; all dense math via V_WMMA_F32_16X16X4_F32 (full f32 precision).
// ---------------------------------------------------------------------------

#define NN    1024
#define CC    15
#define CP    16      // padded C
#define EE    16384
#define LL    3
#define HH    256
#define BB    512

typedef __attribute__((ext_vector_type(2))) float v2f;
typedef __attribute__((ext_vector_type(8))) float v8f;

__device__ __forceinline__ v8f wmma_f32_16x16x4(v2f a, v2f b, v8f c) {
  // 8 args: (neg_a, A, neg_b, B, c_mod, C, reuse_a, reuse_b)
  return __builtin_amdgcn_wmma_f32_16x16x4_f32(false, a, false, b, (short)0, c,
                                               false, false);
}

__device__ __forceinline__ float sigmoidf_(float x) {
  return 1.0f / (1.0f + expf(-x));
}

// ============================ Kernel 1 =====================================
// One block (256 threads = 8 waves) per graph.
__global__ __launch_bounds__(256) void ggnn_graph_kernel(
    const float* __restrict__ X,      // [B,N,C]
    const int*   __restrict__ A,      // [B,2,E]
    const float* __restrict__ V,      // [B,E]
    const float* __restrict__ Wmsg,   // [L,C,C]
    const float* __restrict__ Wih,    // [3C,C]
    const float* __restrict__ Whh,    // [3C,C]
    const float* __restrict__ bih,    // [3C]
    const float* __restrict__ bhh,    // [3C]
    float* __restrict__ pooled)       // [B,16] scratch out
{
  __shared__ float sh_h  [NN * CP];       // 64 KB  node state (padded col15=0)
  __shared__ float sh_agg[NN * CP];       // 64 KB  scatter accumulator
  __shared__ float sh_wm [LL * CP * CP];  // 3 KB   padded W_msg, [l][k][j]
  __shared__ float sh_wi [CP * 48];       // 3 KB   padded W_ih^T, [k][g*16+c]
  __shared__ float sh_wh [CP * 48];       // 3 KB   padded W_hh^T
  __shared__ float sh_bi [48];
  __shared__ float sh_bh [48];

  const int b    = blockIdx.x;
  const int tid  = threadIdx.x;
  const int lane = tid & 31;
  const int wave = tid >> 5;
  const int mrow  = lane & 15;   // A-row / B-col / D-col index
  const int khalf = lane >> 4;   // selects K pair within a K=4 chunk

  // ---- load node features (pad col 15 with 0) ----
  for (int idx = tid; idx < NN * CP; idx += 256) {
    int n = idx >> 4, c = idx & 15;
    sh_h[idx] = (c < CC) ? X[(size_t)b * NN * CC + n * CC + c] : 0.0f;
  }
  // ---- load padded weights ----
  for (int idx = tid; idx < LL * CP * CP; idx += 256) {
    int l = idx >> 8, rem = idx & 255, i = rem >> 4, j = rem & 15;
    sh_wm[idx] = (i < CC && j < CC) ? Wmsg[l * CC * CC + i * CC + j] : 0.0f;
  }
  for (int idx = tid; idx < CP * 48; idx += 256) {
    int k = idx / 48, col = idx % 48, g = col >> 4, c = col & 15;
    bool ok = (k < CC) && (c < CC);
    sh_wi[idx] = ok ? Wih[(g * CC + c) * CC + k] : 0.0f;
    sh_wh[idx] = ok ? Whh[(g * CC + c) * CC + k] : 0.0f;
  }
  if (tid < 48) {
    int g = tid >> 4, c = tid & 15;
    sh_bi[tid] = (c < CC) ? bih[g * CC + c] : 0.0f;
    sh_bh[tid] = (c < CC) ? bhh[g * CC + c] : 0.0f;
  }

  const int* esrc = A + (size_t)b * 2 * EE;
  const int* edst = esrc + EE;
  const float* ev = V + (size_t)b * EE;

  for (int l = 0; l < LL; ++l) {
    __syncthreads();                       // h complete
    for (int idx = tid; idx < NN * CP; idx += 256) sh_agg[idx] = 0.0f;
    __syncthreads();

    // ---- scatter: agg[dst] += v * h[src]   (linear-rewrite: W applied later)
    for (int e = tid; e < EE; e += 256) {
      int s = esrc[e], d = edst[e];
      float vv = ev[e];
      const float* hs = &sh_h[s * CP];
      float* ad = &sh_agg[d * CP];
      #pragma unroll
      for (int c = 0; c < CC; ++c) atomicAdd(&ad[c], vv * hs[c]);
    }
    __syncthreads();

    // ---- per-wave 16-row tiles: agg <- agg @ Wmsg[l]; then GRU update ----
    for (int t = wave; t < NN / 16; t += 8) {
      const int r0 = t * 16;

      // agg @ Wmsg  (in-place, tile-local)
      v2f aT[4];
      #pragma unroll
      for (int kc = 0; kc < 4; ++kc) {
        int k0 = 4 * kc + 2 * khalf;
        const float* ap = &sh_agg[(r0 + mrow) * CP];
        aT[kc][0] = ap[k0]; aT[kc][1] = ap[k0 + 1];
      }
      v8f mC;
      #pragma unroll
      for (int v = 0; v < 8; ++v) mC[v] = 0.0f;
      #pragma unroll
      for (int kc = 0; kc < 4; ++kc) {
        int kb = 4 * kc + 2 * khalf;
        v2f bf;
        bf[0] = sh_wm[l * 256 + kb * CP + mrow];
        bf[1] = sh_wm[l * 256 + (kb + 1) * CP + mrow];
        mC = wmma_f32_16x16x4(aT[kc], bf, mC);
      }
      #pragma unroll
      for (int v = 0; v < 8; ++v)
        sh_agg[(r0 + v + 8 * khalf) * CP + mrow] = mC[v];

      // reload transformed agg + h as A fragments (same wave: LDS in-order)
      v2f aG[4], aH[4];
      #pragma unroll
      for (int kc = 0; kc < 4; ++kc) {
        int k0 = 4 * kc + 2 * khalf;
        const float* ag = &sh_agg[(r0 + mrow) * CP];
        const float* ah = &sh_h  [(r0 + mrow) * CP];
        aG[kc][0] = ag[k0]; aG[kc][1] = ag[k0 + 1];
        aH[kc][0] = ah[k0]; aH[kc][1] = ah[k0 + 1];
      }

      // GRU gates: gi = agg @ Wih^T + bih, gh = h @ Whh^T + bhh
      v8f gi[3], gh[3];
      #pragma unroll
      for (int g = 0; g < 3; ++g) {
        float bi = sh_bi[g * 16 + mrow];
        float bh = sh_bh[g * 16 + mrow];
        v8f ci, ch;
        #pragma unroll
        for (int v = 0; v < 8; ++v) { ci[v] = bi; ch[v] = bh; }
        #pragma unroll
        for (int kc = 0; kc < 4; ++kc) {
          int kb = 4 * kc + 2 * khalf;
          v2f bfi, bfh;
          bfi[0] = sh_wi[kb * 48 + g * 16 + mrow];
          bfi[1] = sh_wi[(kb + 1) * 48 + g * 16 + mrow];
          bfh[0] = sh_wh[kb * 48 + g * 16 + mrow];
          bfh[1] = sh_wh[(kb + 1) * 48 + g * 16 + mrow];
          ci = wmma_f32_16x16x4(aG[kc], bfi, ci);
          ch = wmma_f32_16x16x4(aH[kc], bfh, ch);
        }
        gi[g] = ci; gh[g] = ch;
      }

      // elementwise GRU, write new h (tile-local, in place)
      #pragma unroll
      for (int v = 0; v < 8; ++v) {
        int rr = r0 + v + 8 * khalf;
        float rg = sigmoidf_(gi[0][v] + gh[0][v]);
        float zg = sigmoidf_(gi[1][v] + gh[1][v]);
        float ng = tanhf(gi[2][v] + rg * gh[2][v]);
        float ho = sh_h[rr * CP + mrow];
        sh_h[rr * CP + mrow] = (1.0f - zg) * ng + zg * ho;
      }
    }
  }
  __syncthreads();

  // ---- mean pool over nodes ----
  float part[CP];
  #pragma unroll
  for (int c = 0; c < CP; ++c) part[c] = 0.0f;
  for (int n = tid; n < NN; n += 256) {
    #pragma unroll
    for (int c = 0; c < CP; ++c) part[c] += sh_h[n * CP + c];
  }
  #pragma unroll
  for (int c = 0; c < CP; ++c) sh_agg[tid * CP + c] = part[c];
  __syncthreads();
  for (int s = 128; s >= 1; s >>= 1) {
    if (tid < s) {
      #pragma unroll
      for (int c = 0; c < CP; ++c)
        sh_agg[tid * CP + c] += sh_agg[(tid + s) * CP + c];
    }
    __syncthreads();
  }
  if (tid < CP)
    pooled[b * CP + tid] =
        (tid < CC) ? sh_agg[tid] * (1.0f / (float)NN) : 0.0f;
}

// ============================ Kernel 2 =====================================
// MLP on pooled features: 32 blocks x 1 wave; 16 graphs per block.
__global__ __launch_bounds__(32) void ggnn_mlp_kernel(
    const float* __restrict__ pooled,  // [B,16]
    const float* __restrict__ W1, const float* __restrict__ b1,  // [15,256],[256]
    const float* __restrict__ W2, const float* __restrict__ b2,  // [256,256],[256]
    const float* __restrict__ W3, const float* __restrict__ b3,  // [256],[1]
    float* __restrict__ out)           // [B]
{
  __shared__ float sW1[CP * HH];   // padded W1 (row 15 = 0)     16 KB
  __shared__ float sh1[16 * HH];   // hidden1 tile               16 KB
  __shared__ float sh2[16 * HH];   // hidden2 tile               16 KB
  __shared__ float spsum[32];

  const int lane  = threadIdx.x;
  const int r0    = blockIdx.x * 16;
  const int mrow  = lane & 15;
  const int khalf = lane >> 4;

  for (int idx = lane; idx < CP * HH; idx += 32) {
    int k = idx >> 8, n = idx & 255;
    sW1[idx] = (k < CC) ? W1[k * HH + n] : 0.0f;
  }
  __syncthreads();

  // ---- stage 1: h1 = relu(pooled @ W1 + b1) ----
  v2f aP[4];
  #pragma unroll
  for (int kc = 0; kc < 4; ++kc) {
    int k0 = 4 * kc + 2 * khalf;
    const float* pr = &pooled[(r0 + mrow) * CP];
    aP[kc][0] = pr[k0]; aP[kc][1] = pr[k0 + 1];
  }
  for (int j = 0; j < 16; ++j) {
    float bb = b1[j * 16 + mrow];
    v8f c;
    #pragma unroll
    for (int v = 0; v < 8; ++v) c[v] = bb;
    #pragma unroll
    for (int kc = 0; kc < 4; ++kc) {
      int kb = 4 * kc + 2 * khalf;
      v2f bf;
      bf[0] = sW1[kb * HH + j * 16 + mrow];
      bf[1] = sW1[(kb + 1) * HH + j * 16 + mrow];
      c = wmma_f32_16x16x4(aP[kc], bf, c);
    }
    #pragma unroll
    for (int v = 0; v < 8; ++v)
      sh1[(v + 8 * khalf) * HH + j * 16 + mrow] = fmaxf(c[v], 0.0f);
  }
  __syncthreads();

  // ---- stage 2: h2 = relu(h1 @ W2 + b2) ----
  for (int j = 0; j < 16; ++j) {
    float bb = b2[j * 16 + mrow];
    v8f c;
    #pragma unroll
    for (int v = 0; v < 8; ++v) c[v] = bb;
    for (int kc = 0; kc < HH / 4; ++kc) {
      int kb = 4 * kc + 2 * khalf;
      v2f af, bf;
      af[0] = sh1[mrow * HH + kb];
      af[1] = sh1[mrow * HH + kb + 1];
      bf[0] = W2[kb * HH + j * 16 + mrow];
      bf[1] = W2[(kb + 1) * HH + j * 16 + mrow];
      c = wmma_f32_16x16x4(af, bf, c);
    }
    #pragma unroll
    for (int v = 0; v < 8; ++v)
      sh2[(v + 8 * khalf) * HH + j * 16 + mrow] = fmaxf(c[v], 0.0f);
  }
  __syncthreads();

  // ---- stage 3: out = h2 @ W3 + b3 (two lanes per row) ----
  float p = 0.0f;
  for (int k = 0; k < HH / 2; ++k) {
    int kk = khalf * (HH / 2) + k;
    p += sh2[mrow * HH + kk] * W3[kk];
  }
  spsum[lane] = p;
  __syncthreads();
  if (lane < 16) out[r0 + lane] = spsum[lane] + spsum[lane + 16] + b3[0];
}

// ============================ Launch =======================================
extern "C" void kernel_launch(void* const* d_in, const int* in_sizes, int n_in,
                              void* d_out, int out_size, void* d_ws, size_t ws_size,
                              hipStream_t stream) {
  const float* X    = (const float*)d_in[0];
  const int*   A    = (const int*)  d_in[1];
  const float* V    = (const float*)d_in[2];
  const float* Wmsg = (const float*)d_in[3];
  const float* Wih  = (const float*)d_in[4];
  const float* Whh  = (const float*)d_in[5];
  const float* bih  = (const float*)d_in[6];
  const float* bhh  = (const float*)d_in[7];
  const float* W1   = (const float*)d_in[8];
  const float* b1   = (const float*)d_in[9];
  const float* W2   = (const float*)d_in[10];
  const float* b2   = (const float*)d_in[11];
  const float* W3   = (const float*)d_in[12];
  const float* b3   = (const float*)d_in[13];
  float* out    = (float*)d_out;
  float* pooled = (float*)d_ws;   // [512*16] f32 = 32 KB

  ggnn_graph_kernel<<<dim3(BB), dim3(256), 0, stream>>>(
      X, A, V, Wmsg, Wih, Whh, bih, bhh, pooled);
  ggnn_mlp_kernel<<<dim3(BB / 16), dim3(32), 0, stream>>>(
      pooled, W1, b1, W2, b2, W3, b3, out);
}